// LoRARelationWiseWordSelectionHead_69836168233630
// MI455X (gfx1250) — compile-verified
//
#include <hip/hip_runtime.h>
#include <math.h>

typedef __attribute__((ext_vector_type(2))) float v2f;
typedef __attribute__((ext_vector_type(8))) float v8f;

#define H 768
#define LBL 8
#define BS 192   // b*s = 2*96
#define SEQ 96

// ---------------------------------------------------------------------------
// Phase 2: src_all[bs][l][h] = pooled[bs][:] @ (W.T + da@db[:,:,l]) + bias
// One block = (l, ntile, sel). 128 threads = 4 waves; wave w computes m-tiles
// {w, w+4, w+8} (16 rows each) of the 192-row GEMM using V_WMMA_F32_16X16X4_F32.
// Effective weight tile (768 x 16) built cooperatively into LDS first.
// ---------------------------------------------------------------------------
__global__ __launch_bounds__(128) void lora_proj_wmma(
    const float* __restrict__ pooled,
    const float* __restrict__ W_src, const float* __restrict__ b_src,
    const float* __restrict__ W_tgt, const float* __restrict__ b_tgt,
    const float* __restrict__ da_src, const float* __restrict__ db_src,
    const float* __restrict__ da_tgt, const float* __restrict__ db_tgt,
    float* __restrict__ srcAll, float* __restrict__ tgtAll)
{
    const int ntile = blockIdx.x;        // 0..47
    const int l     = blockIdx.y;        // 0..7
    const int sel   = blockIdx.z;        // 0 = src, 1 = tgt
    const int nbase = ntile * 16;

    const float* __restrict__ W    = sel ? W_tgt  : W_src;
    const float* __restrict__ da   = sel ? da_tgt : da_src;
    const float* __restrict__ db   = sel ? db_tgt : db_src;
    const float* __restrict__ bias = sel ? b_tgt  : b_src;
    float* __restrict__ dst        = sel ? tgtAll : srcAll;

    // layout: effT[(k>>1)*32 + n*2 + (k&1)]  -> (kk, kk+1) pair contiguous,
    // conflict-free across the 32 lanes of a wave.
    __shared__ float effT[H * 16];

    for (int idx = threadIdx.x; idx < 16 * H; idx += 128) {
        int n = idx / H;                 // 0..15 (local output column)
        int k = idx % H;                 // input feature
        float v = W[(nbase + n) * H + k];          // W.T[k][n]
#pragma unroll
        for (int r = 0; r < 4; ++r)
            v += da[(k * 4 + r) * LBL + l] * db[(r * H + nbase + n) * LBL + l];
        effT[(k >> 1) * 32 + n * 2 + (k & 1)] = v;
    }
    __syncthreads();

    const int wave = threadIdx.x >> 5;   // 0..3
    const int lane = threadIdx.x & 31;
    const int nl   = lane & 15;          // N column (and A-matrix row)
    const int half = lane >> 4;          // 0: K=0,1   1: K=2,3

    const float bval = bias[nbase + nl];

    const int m0 = (wave + 0) * 16;
    const int m1 = (wave + 4) * 16;
    const int m2 = (wave + 8) * 16;

    v8f acc0 = {}; v8f acc1 = {}; v8f acc2 = {};

    for (int k = 0; k < H; k += 4) {
        const int kk = k + 2 * half;     // this lane's K pair
        // B fragment (4x16 f32): rows kk, kk+1 at column nl
        const v2f bfrag = *(const v2f*)(&effT[(kk >> 1) * 32 + nl * 2]);
        // A fragments (16x4 f32): row nl, K = kk, kk+1
        const v2f a0 = *(const v2f*)(pooled + (m0 + nl) * H + kk);
        const v2f a1 = *(const v2f*)(pooled + (m1 + nl) * H + kk);
        const v2f a2 = *(const v2f*)(pooled + (m2 + nl) * H + kk);
        acc0 = __builtin_amdgcn_wmma_f32_16x16x4_f32(false, a0, false, bfrag,
                                                     (short)0, acc0, false, false);
        acc1 = __builtin_amdgcn_wmma_f32_16x16x4_f32(false, a1, false, bfrag,
                                                     (short)0, acc1, false, false);
        acc2 = __builtin_amdgcn_wmma_f32_16x16x4_f32(false, a2, false, bfrag,
                                                     (short)0, acc2, false, false);
    }

    // C/D layout: reg i -> row (i + 8*half), col nl. dst layout [bs][l][h].
#pragma unroll
    for (int i = 0; i < 8; ++i) {
        const int mrow = i + 8 * half;
        const long col = (long)l * H + nbase + nl;
        dst[(long)(m0 + mrow) * (LBL * H) + col] = acc0[i] + bval;
        dst[(long)(m1 + mrow) * (LBL * H) + col] = acc1[i] + bval;
        dst[(long)(m2 + mrow) * (LBL * H) + col] = acc2[i] + bval;
    }
}

// ---------------------------------------------------------------------------
// Phase 3: out[b,s,t,l] = sum_h gelu(src[b,s,l,h] + tgt[b,t,l,h]) * cls[h,l]
// Block = (b, l, 16x16 (s,t) tile). Activation tiles staged in LDS (row pad
// 772 floats -> conflict-free), exact GELU via erff, 768-deep accumulation.
// ---------------------------------------------------------------------------
__global__ __launch_bounds__(256) void pair_gelu_reduce(
    const float* __restrict__ srcAll, const float* __restrict__ tgtAll,
    const float* __restrict__ classifier, float* __restrict__ out)
{
    const int tile  = blockIdx.x;        // 0..35
    const int stile = tile / 6;
    const int ttile = tile % 6;
    const int l     = blockIdx.y;        // 0..7
    const int b     = blockIdx.z;        // 0..1

    __shared__ float sT[16 * 772];
    __shared__ float tT[16 * 772];
    __shared__ float cl[H];

    for (int i = threadIdx.x; i < 16 * (H / 4); i += 256) {
        const int row = i / (H / 4);
        const int q   = i % (H / 4);
        const long sOff = ((long)(b * SEQ + stile * 16 + row) * LBL + l) * H + q * 4;
        const long tOff = ((long)(b * SEQ + ttile * 16 + row) * LBL + l) * H + q * 4;
        *(float4*)&sT[row * 772 + q * 4] = *(const float4*)(srcAll + sOff);
        *(float4*)&tT[row * 772 + q * 4] = *(const float4*)(tgtAll + tOff);
    }
    for (int i = threadIdx.x; i < H; i += 256)
        cl[i] = classifier[i * LBL + l];
    __syncthreads();

    const int sl = threadIdx.x & 15;
    const int tl = threadIdx.x >> 4;
    const float* __restrict__ sp = &sT[sl * 772];
    const float* __restrict__ tp = &tT[tl * 772];

    float acc = 0.0f;
#pragma unroll 4
    for (int h = 0; h < H; ++h) {
        const float x = sp[h] + tp[h];
        const float g = 0.5f * x * (1.0f + erff(x * 0.70710678118654752f));
        acc = fmaf(g, cl[h], acc);
    }

    const int s = stile * 16 + sl;
    const int t = ttile * 16 + tl;
    out[(((long)b * SEQ + s) * SEQ + t) * LBL + l] = acc;
}

// ---------------------------------------------------------------------------
extern "C" void kernel_launch(void* const* d_in, const int* in_sizes, int n_in,
                              void* d_out, int out_size, void* d_ws, size_t ws_size,
                              hipStream_t stream) {
    const float* pooled = (const float*)d_in[0];
    const float* W_src  = (const float*)d_in[1];
    const float* b_src  = (const float*)d_in[2];
    const float* W_tgt  = (const float*)d_in[3];
    const float* b_tgt  = (const float*)d_in[4];
    const float* da_src = (const float*)d_in[5];
    const float* db_src = (const float*)d_in[6];
    const float* da_tgt = (const float*)d_in[7];
    const float* db_tgt = (const float*)d_in[8];
    const float* cls    = (const float*)d_in[9];

    float* srcAll = (float*)d_ws;                       // [192][8][768]
    float* tgtAll = srcAll + (size_t)BS * LBL * H;      // [192][8][768]

    dim3 g2(48, LBL, 2);
    lora_proj_wmma<<<g2, 128, 0, stream>>>(pooled, W_src, b_src, W_tgt, b_tgt,
                                           da_src, db_src, da_tgt, db_tgt,
                                           srcAll, tgtAll);

    dim3 g3(36, LBL, 2);
    pair_gelu_reduce<<<g3, 256, 0, stream>>>(srcAll, tgtAll, cls, (float*)d_out);
}